// transformer_38723425141266
// MI455X (gfx1250) — compile-verified
//
#include <hip/hip_runtime.h>
#include <math.h>

#define B_  2
#define S_  1024
#define D_  512
#define H_  32
#define DH_ 16
#define L_  4
#define M_  (B_*S_)          // 2048 rows
#define MD_ ((size_t)M_*D_)  // elements per activation buffer

typedef float v2f __attribute__((ext_vector_type(2)));
typedef float v8f __attribute__((ext_vector_type(8)));

static __device__ __forceinline__ v8f wmma4(v2f a, v2f b, v8f c) {
  // D = A(16x4 f32) * B(4x16 f32) + C(16x16 f32), full fp32 accuracy.
  return __builtin_amdgcn_wmma_f32_16x16x4_f32(false, a, false, b, (short)0, c, false, false);
}

static __device__ __forceinline__ float gelu_exact(float x) {
  return 0.5f * x * (1.0f + erff(x * 0.7071067811865475f));
}

// h = gelu(x[:,None]*Winit + binit)  -> [M_, D_]
__global__ void init_kernel(const float* __restrict__ x, const float* __restrict__ Winit,
                            const float* __restrict__ binit, float* __restrict__ h) {
  size_t i = (size_t)blockIdx.x * blockDim.x + threadIdx.x;
  if (i >= MD_) return;
  int col = (int)(i & (D_ - 1));
  float val = x[i >> 9] * Winit[col] + binit[col];   // D_ == 512 == 1<<9
  h[i] = gelu_exact(val);
}

// C[M,N] = A[M,K] @ W[K,N] + bias[N]  (row-major fp32, fp32 WMMA 16x16x4)
// 256 threads = 8 waves; wave w -> 16(M) x 64(N) tile; block tile 128x64.
// W staged through double-buffered LDS, column-major so a B-fragment
// (two consecutive K-rows, one column) is a single aligned ds_load_b64.
// Grid: (M/128, N/64). Dimensions are exact multiples (2048, 512, 512).
__global__ void gemm_bias_kernel(const float* __restrict__ A, const float* __restrict__ W,
                                 const float* __restrict__ bias, float* __restrict__ C,
                                 int Mdim, int Ndim, int Kdim) {
  __shared__ alignas(16) float Wt[2][64][18];   // [buf][col 0..63][krow 0..15], pad 18

  const int tid  = threadIdx.x;
  const int lane = tid & 31;
  const int wave = tid >> 5;
  const int g  = lane >> 4;   // lane half
  const int ln = lane & 15;
  const int m0 = blockIdx.x * 128 + wave * 16;
  const int n0 = blockIdx.y * 64;

  // staging role: thread covers column sc, k-rows sr, sr+4, sr+8, sr+12
  const int sc = tid & 63;
  const int sr = tid >> 6;          // 0..3
  const float* wstage = W + (size_t)sr * Ndim + n0 + sc;

  v8f acc[4];
  #pragma unroll
  for (int j = 0; j < 4; ++j) acc[j] = (v8f){0,0,0,0,0,0,0,0};

  const float* arow = A + (size_t)(m0 + ln) * Kdim + 2 * g;
  const int nslab = Kdim >> 4;      // 16-wide K slabs

  // preload slab 0
  #pragma unroll
  for (int i = 0; i < 4; ++i)
    Wt[0][sc][sr + 4 * i] = wstage[(size_t)(4 * i) * Ndim];
  __syncthreads();

  #pragma unroll 2
  for (int ks = 0; ks < nslab; ++ks) {
    const int buf = ks & 1;
    if (ks + 1 < nslab) {
      const float* wnext = wstage + (size_t)(ks + 1) * 16 * Ndim;
      #pragma unroll
      for (int i = 0; i < 4; ++i)
        Wt[buf ^ 1][sc][sr + 4 * i] = wnext[(size_t)(4 * i) * Ndim];
    }
    const float* ap = arow + ks * 16;
    #pragma unroll
    for (int j4 = 0; j4 < 4; ++j4) {
      v2f a = *(const v2f*)(ap + 4 * j4);                 // A[m][ks*16+4*j4+2g .. +1]
      #pragma unroll
      for (int j = 0; j < 4; ++j) {
        // B reg pair = W[k4+2g][n], W[k4+2g+1][n]: contiguous in column-major LDS
        v2f b = *(const v2f*)(&Wt[buf][j * 16 + ln][4 * j4 + 2 * g]);
        acc[j] = wmma4(a, b, acc[j]);
      }
    }
    __syncthreads();
  }

  #pragma unroll
  for (int j = 0; j < 4; ++j) {
    int n = n0 + j * 16 + ln;
    float bval = bias[n];
    #pragma unroll
    for (int v = 0; v < 8; ++v) {
      int m = m0 + v + 8 * g;           // C/D layout: row = v + 8*(l/16), col = l%16
      C[(size_t)m * Ndim + n] = acc[j][v] + bval;
    }
  }
}

// Flash-style causal attention with relative-position (skew) term.
// One wave (32 thr) per (b, head, 16-row query tile). grid = (B*H, S/16).
__global__ void attn_kernel(const float* __restrict__ Q, const float* __restrict__ K,
                            const float* __restrict__ V, const float* __restrict__ Er,
                            float* __restrict__ O) {
  __shared__ alignas(16) float qs[16][16];
  __shared__ alignas(16) float er[32][16];
  __shared__ alignas(16) float ps[16][16];

  const int lane = threadIdx.x;  // 0..31
  const int g  = lane >> 4;
  const int ln = lane & 15;
  const int bh = blockIdx.x;
  const int b  = bh >> 5;        // / H_
  const int hh = bh & 31;
  const int s0 = blockIdx.y * 16;
  const int hbase = hh * DH_;
  const size_t rowb = (size_t)b * S_;

  // stage 16x16 q tile
  #pragma unroll
  for (int i = 0; i < 8; ++i) {
    int e = lane + 32 * i;
    int r = e >> 4, c = e & 15;
    qs[r][c] = Q[(rowb + s0 + r) * D_ + hbase + c];
  }
  __syncthreads();

  float mrow[8], lrow[8];
  v8f o = (v8f){0,0,0,0,0,0,0,0};
  #pragma unroll
  for (int v = 0; v < 8; ++v) { mrow[v] = -3.0e38f; lrow[v] = 0.0f; }

  const int ntile = (s0 >> 4) + 1;   // causal: t0 <= s0
  for (int tt = 0; tt < ntile; ++tt) {
    const int t0 = tt * 16;
    // stage Er diagonal band: rows rbase .. rbase+31  (r = S-1-s+t)
    const int rbase = S_ - 16 - s0 + t0;
    #pragma unroll
    for (int i = 0; i < 16; ++i) {
      int e = lane + 32 * i;
      int br = e >> 4, c = e & 15;
      int r = rbase + br;
      er[br][c] = (r >= 0 && r < S_) ? Er[r * DH_ + c] : 0.0f;
    }
    __syncthreads();

    // scores tile = q @ k^T via 4 fp32 WMMAs (K-dim = DH = 16)
    v8f acc = (v8f){0,0,0,0,0,0,0,0};
    #pragma unroll
    for (int j = 0; j < 4; ++j) {
      int kk = 4 * j;
      v2f a  = *(const v2f*)(&qs[ln][kk + 2 * g]);
      v2f bb = *(const v2f*)(&K[(rowb + t0 + ln) * D_ + hbase + kk + 2 * g]);
      acc = wmma4(a, bb, acc);
    }

    // + Srel, scale, causal mask; online softmax
    float p[8], tmax[8];
    #pragma unroll
    for (int v = 0; v < 8; ++v) {
      int si = v + 8 * g, ti = ln;
      int s = s0 + si, t = t0 + ti;
      float sc;
      if (t <= s) {
        int bi = 15 - si + ti;            // er row: rbase+bi == S-1-s+t
        float srel = 0.0f;
        #pragma unroll
        for (int c = 0; c < 16; ++c) srel += qs[si][c] * er[bi][c];
        sc = (acc[v] + srel) * 0.25f;     // 1/sqrt(DH) = 0.25
      } else {
        sc = -3.0e38f;
      }
      p[v] = sc;
      tmax[v] = sc;
    }
    // row-wise max (each row lives in one 16-lane half at fixed reg v)
    #pragma unroll
    for (int off = 1; off < 16; off <<= 1) {
      #pragma unroll
      for (int v = 0; v < 8; ++v)
        tmax[v] = fmaxf(tmax[v], __shfl_xor(tmax[v], off, 32));
    }
    float rsum[8];
    #pragma unroll
    for (int v = 0; v < 8; ++v) {
      float mnew  = fmaxf(mrow[v], tmax[v]);
      float scale = expf(mrow[v] - mnew);
      p[v] = expf(p[v] - mnew);
      mrow[v] = mnew;
      lrow[v] *= scale;
      o[v]    *= scale;
      rsum[v] = p[v];
    }
    #pragma unroll
    for (int off = 1; off < 16; off <<= 1) {
      #pragma unroll
      for (int v = 0; v < 8; ++v) rsum[v] += __shfl_xor(rsum[v], off, 32);
    }
    #pragma unroll
    for (int v = 0; v < 8; ++v) lrow[v] += rsum[v];

    // reshape P (C-layout) -> A-fragments through LDS, then O += P @ V
    #pragma unroll
    for (int v = 0; v < 8; ++v) ps[v + 8 * g][ln] = p[v];
    __syncthreads();
    #pragma unroll
    for (int j = 0; j < 4; ++j) {
      int kk = 4 * j;
      v2f a = *(const v2f*)(&ps[ln][kk + 2 * g]);
      v2f bb;
      bb.x = V[(rowb + t0 + kk + 2 * g    ) * D_ + hbase + ln];
      bb.y = V[(rowb + t0 + kk + 2 * g + 1) * D_ + hbase + ln];
      o = wmma4(a, bb, o);
    }
    __syncthreads();
  }

  #pragma unroll
  for (int v = 0; v < 8; ++v) {
    int si = v + 8 * g;
    O[(rowb + s0 + si) * D_ + hbase + ln] = o[v] / lrow[v];
  }
}

// h += gelu(layernorm(Y_row) * gamma + beta); one block (256 thr) per row.
__global__ void ln_gelu_res_kernel(const float* __restrict__ Y, const float* __restrict__ gamma,
                                   const float* __restrict__ beta, float* __restrict__ h) {
  __shared__ float red[256];
  const int row = blockIdx.x;
  const int tid = threadIdx.x;
  const float* yr = Y + (size_t)row * D_;
  float x0 = yr[tid], x1 = yr[tid + 256];
  red[tid] = x0 + x1;
  __syncthreads();
  for (int s = 128; s > 0; s >>= 1) { if (tid < s) red[tid] += red[tid + s]; __syncthreads(); }
  float mu = red[0] * (1.0f / D_);
  __syncthreads();
  float d0 = x0 - mu, d1 = x1 - mu;
  red[tid] = d0 * d0 + d1 * d1;
  __syncthreads();
  for (int s = 128; s > 0; s >>= 1) { if (tid < s) red[tid] += red[tid + s]; __syncthreads(); }
  float var = red[0] * (1.0f / D_);
  float rs = rsqrtf(var + 1e-5f);
  float y0 = d0 * rs * gamma[tid]       + beta[tid];
  float y1 = d1 * rs * gamma[tid + 256] + beta[tid + 256];
  float* hr = h + (size_t)row * D_;
  hr[tid]       += gelu_exact(y0);
  hr[tid + 256] += gelu_exact(y1);
}

// out[b] = (sum_s h[b,s,:]) @ Wout
__global__ void out_kernel(const float* __restrict__ h, const float* __restrict__ Wout,
                           float* __restrict__ out) {
  __shared__ float red[256];
  const int b = blockIdx.x;
  const int tid = threadIdx.x;
  float partial = 0.0f;
  for (int d = tid; d < D_; d += 256) {
    float cs = 0.0f;
    const float* col = h + (size_t)b * S_ * D_ + d;
    for (int s = 0; s < S_; ++s) cs += col[(size_t)s * D_];
    partial += cs * Wout[d];
  }
  red[tid] = partial;
  __syncthreads();
  for (int s = 128; s > 0; s >>= 1) { if (tid < s) red[tid] += red[tid + s]; __syncthreads(); }
  if (tid == 0) out[b] = red[0];
}

extern "C" void kernel_launch(void* const* d_in, const int* in_sizes, int n_in,
                              void* d_out, int out_size, void* d_ws, size_t ws_size,
                              hipStream_t stream) {
  const float* x     = (const float*)d_in[0];
  const float* Winit = (const float*)d_in[1];
  const float* binit = (const float*)d_in[2];
  const float* Wq    = (const float*)d_in[3];
  const float* bq    = (const float*)d_in[4];
  const float* Wk    = (const float*)d_in[5];
  const float* bk    = (const float*)d_in[6];
  const float* Wv    = (const float*)d_in[7];
  const float* bv    = (const float*)d_in[8];
  const float* Er    = (const float*)d_in[9];
  const float* Wlin  = (const float*)d_in[10];
  const float* blin  = (const float*)d_in[11];
  const float* ln_g  = (const float*)d_in[12];
  const float* ln_b  = (const float*)d_in[13];
  const float* Wout  = (const float*)d_in[14];
  float* out = (float*)d_out;

  float* ws   = (float*)d_ws;
  float* h    = ws;
  float* q    = ws + 1 * MD_;
  float* k    = ws + 2 * MD_;
  float* v    = ws + 3 * MD_;
  float* attn = ws + 4 * MD_;
  float* yraw = q;   // q dead after attention; reuse for lin-GEMM output

  init_kernel<<<(int)(MD_ / 256), 256, 0, stream>>>(x, Winit, binit, h);

  dim3 ggrid(M_ / 128, D_ / 64);
  dim3 agrid(B_ * H_, S_ / 16);
  for (int l = 0; l < L_; ++l) {
    const size_t wOff = (size_t)l * D_ * D_;
    gemm_bias_kernel<<<ggrid, 256, 0, stream>>>(h, Wq + wOff, bq + l * D_, q, M_, D_, D_);
    gemm_bias_kernel<<<ggrid, 256, 0, stream>>>(h, Wk + wOff, bk + l * D_, k, M_, D_, D_);
    gemm_bias_kernel<<<ggrid, 256, 0, stream>>>(h, Wv + wOff, bv + l * D_, v, M_, D_, D_);
    attn_kernel<<<agrid, 32, 0, stream>>>(q, k, v, Er + (size_t)l * S_ * DH_, attn);
    gemm_bias_kernel<<<ggrid, 256, 0, stream>>>(attn, Wlin + wOff, blin + l * D_, yraw, M_, D_, D_);
    ln_gelu_res_kernel<<<M_, 256, 0, stream>>>(yraw, ln_g + l * D_, ln_b + l * D_, h);
  }
  out_kernel<<<B_, 256, 0, stream>>>(h, Wout, out);
}